// Weight_2px_Pad2d_11742440587594
// MI455X (gfx1250) — compile-verified
//
#include <hip/hip_runtime.h>

// Problem constants (reference: B_IMG=32, P=2, C=128, H=56, W=56; b = 32*2*2 = 128)
#define C_CH      128
#define H_IN      56
#define W_IN      56
#define H_OUT     58
#define W_OUT     58
#define PLANE_IN  (H_IN * W_IN)    // 3136 floats (12544 B, 16B-aligned per plane)
#define PLANE_OUT (H_OUT * W_OUT)  // 3364 floats (13456 B, 16B-aligned per plane)
#define NVEC_IN   (PLANE_IN / 4)   // 784 float4 per input plane (row = 14 vec4 exactly)
#define NVEC_OUT  ((PLANE_OUT + 3) / 4) // 841 float4 per output plane
#define NT        256

// Truncate a generic pointer to the 32-bit wave-relative LDS byte address
// (CDNA5 flat aperture: LDS address is the low 32 bits of the flat address).
__device__ __forceinline__ unsigned lds_b32_addr(const void* p) {
    return (unsigned)(unsigned long long)p;
}

__global__ __launch_bounds__(NT) void weight_pad2d_kernel(
    const float* __restrict__ x,
    const float* __restrict__ topW,  const float* __restrict__ botW,
    const float* __restrict__ leftW, const float* __restrict__ rightW,
    const float* __restrict__ tlW,   const float* __restrict__ trW,
    const float* __restrict__ blW,   const float* __restrict__ brW,
    const int*   __restrict__ num_patches,
    float* __restrict__ out)
{
    // Full padded output image for this (batch, channel) plane, built in LDS.
    __shared__ __align__(16) float tile[PLANE_OUT];   // 13456 B

    const int plane = blockIdx.x;          // plane = bi * C + c
    const int c     = plane % C_CH;
    const int bi    = plane / C_CH;
    const int t     = threadIdx.x;

    const float*  in_p  = x   + (size_t)plane * PLANE_IN;
    const float4* in4   = reinterpret_cast<const float4*>(in_p);
    float*        out_p = out + (size_t)plane * PLANE_OUT;

    // ---------- Phase 1: interior, global b128 reads -> shifted LDS writes ----------
    // Input row = exactly 14 float4s, so no vec4 spans two rows.
    #pragma unroll
    for (int i = 0; i < 4; ++i) {
        int q = t + i * NT;
        if (q > NVEC_IN - 1) q = NVEC_IN - 1;   // clamp tail: EXEC stays all-ones
        const float4 v = in4[q];
        const int r    = q / 14;
        const int j    = q % 14;
        const int base = (r + 1) * W_OUT + 4 * j + 1;  // +1 row, +1 col shift
        tile[base + 0] = v.x;
        tile[base + 1] = v.y;
        tile[base + 2] = v.z;
        tile[base + 3] = v.w;
    }

    // ---------- Phase 2: boundary ring (228 elements) computed into LDS ----------
    if (t < 2 * W_IN + 2 * H_IN + 4) {
        const int  P  = *num_patches;            // = 2
        const int  PP = P * P;
        const int  within  = bi % PP;
        const bool top_m   = within < P;
        const bool bot_m   = within >= PP - P;
        const bool left_m  = (bi % P) == 0;
        const bool right_m = (bi % P) == (P - 1);

        float v;
        int   oidx;
        if (t < W_IN) {                                   // top row, col t+1
            const int cc = t;
            v = top_m ? 0.0f : topW[c] * (in_p[cc] + in_p[W_IN + cc]);
            oidx = cc + 1;
        } else if (t < 2 * W_IN) {                        // bottom row
            const int cc = t - W_IN;
            v = bot_m ? 0.0f
                      : botW[c] * (in_p[(H_IN - 2) * W_IN + cc] + in_p[(H_IN - 1) * W_IN + cc]);
            oidx = (H_OUT - 1) * W_OUT + (cc + 1);
        } else if (t < 2 * W_IN + H_IN) {                 // left col, row r+1
            const int r = t - 2 * W_IN;
            v = left_m ? 0.0f : leftW[c] * (in_p[r * W_IN] + in_p[r * W_IN + 1]);
            oidx = (r + 1) * W_OUT;
        } else if (t < 2 * W_IN + 2 * H_IN) {             // right col
            const int r = t - 2 * W_IN - H_IN;
            v = right_m ? 0.0f
                        : rightW[c] * (in_p[r * W_IN + W_IN - 2] + in_p[r * W_IN + W_IN - 1]);
            oidx = (r + 1) * W_OUT + (W_OUT - 1);
        } else {                                          // 4 corners
            const int e = t - (2 * W_IN + 2 * H_IN);
            if (e == 0) {
                v = (top_m || left_m)  ? 0.0f : tlW[c] * in_p[0];
                oidx = 0;
            } else if (e == 1) {
                v = (top_m || right_m) ? 0.0f : trW[c] * in_p[W_IN - 1];
                oidx = W_OUT - 1;
            } else if (e == 2) {
                v = (bot_m || left_m)  ? 0.0f : blW[c] * in_p[(H_IN - 1) * W_IN];
                oidx = (H_OUT - 1) * W_OUT;
            } else {
                v = (bot_m || right_m) ? 0.0f : brW[c] * in_p[H_IN * W_IN - 1];
                oidx = H_OUT * W_OUT - 1;
            }
        }
        tile[oidx] = v;
    }

    // All 3364 LDS floats written (interior + ring) -> make visible across waves.
    __syncthreads();

    // ---------- Phase 3: whole padded plane, LDS -> global via async b128 DMA ----------
    // Output plane is contiguous and 16B-aligned; 841 vec4 transfers.
    #pragma unroll
    for (int i = 0; i < 4; ++i) {
        int o = t + i * NT;
        if (o > NVEC_OUT - 1) o = NVEC_OUT - 1;  // clamp tail: benign duplicate store
        unsigned           ldsa = lds_b32_addr(&tile[4 * o]);           // 16B-aligned
        unsigned long long ga   = (unsigned long long)(const void*)(out_p + 4 * o);
        asm volatile("global_store_async_from_lds_b128 %0, %1, off"
                     :: "v"(ga), "v"(ldsa) : "memory");
    }
    // s_endpgm performs an implicit wait-idle, draining the trailing async stores.
}

extern "C" void kernel_launch(void* const* d_in, const int* in_sizes, int n_in,
                              void* d_out, int out_size, void* d_ws, size_t ws_size,
                              hipStream_t stream) {
    const float* x      = (const float*)d_in[0];
    const float* topW   = (const float*)d_in[1];
    const float* botW   = (const float*)d_in[2];
    const float* leftW  = (const float*)d_in[3];
    const float* rightW = (const float*)d_in[4];
    const float* tlW    = (const float*)d_in[5];
    const float* trW    = (const float*)d_in[6];
    const float* blW    = (const float*)d_in[7];
    const float* brW    = (const float*)d_in[8];
    // d_in[9] = padding (fixed to 1), d_in[10] = num_patches
    const int* num_patches = (const int*)d_in[10];

    const int planes = in_sizes[0] / PLANE_IN;  // b * C = 16384

    weight_pad2d_kernel<<<planes, NT, 0, stream>>>(
        x, topW, botW, leftW, rightW, tlW, trW, blW, brW, num_patches,
        (float*)d_out);
}